// GraphSAGE_40578851013001
// MI455X (gfx1250) — compile-verified
//
#include <hip/hip_runtime.h>

#define N_NODES 100000
#define N_EDGES 640000
#define IN_F 64
#define HID_F 128
#define OUT_F 121

typedef __attribute__((ext_vector_type(2))) float v2f;
typedef __attribute__((ext_vector_type(8))) float v8f;

// ---------------- zero scratch ----------------
__global__ void zero_f32(float* __restrict__ p, int n) {
    int i = blockIdx.x * blockDim.x + threadIdx.x;
    int stride = gridDim.x * blockDim.x;
    for (; i < n; i += stride) p[i] = 0.0f;
}

// ---------------- degree: deg[dst[e]] += 1 ----------------
__global__ void degree_kernel(const long long* __restrict__ dst,
                              float* __restrict__ deg, int nE) {
    int e = blockIdx.x * blockDim.x + threadIdx.x;
    if (e < nE) unsafeAtomicAdd(deg + (int)dst[e], 1.0f);
}

// ---------------- scatter add: agg[dst[e], f] += feat[src[e], f] ----------------
// L2-resident f32 atomics (agg fits in 192MB L2). Coalesced along feature dim.
template <int LOGF>
__global__ void scatter_add(const float* __restrict__ feat,
                            const long long* __restrict__ src,
                            const long long* __restrict__ dst,
                            float* __restrict__ agg, int nE) {
    const int F = 1 << LOGF;
    int tid = blockIdx.x * blockDim.x + threadIdx.x;
    int e = tid >> LOGF;
    int f = tid & (F - 1);
    if (e < nE) {
        int s = (int)src[e];
        int d = (int)dst[e];
        unsafeAtomicAdd(agg + (size_t)d * F + f, feat[(size_t)s * F + f]);
    }
}

// ---------------- layer 1: h = relu(mean@W1l.T + x@W1r.T + b1) ----------------
// One wave -> one 16x16 output tile. fp32 WMMA 16x16x4, K swept in steps of 4.
__global__ void __launch_bounds__(256) sage_layer1(
    const float* __restrict__ x, const float* __restrict__ agg,
    const float* __restrict__ deg, const float* __restrict__ W1l,
    const float* __restrict__ W1r, const float* __restrict__ b1,
    float* __restrict__ h) {
    const int lane = threadIdx.x & 31;
    const int tile = blockIdx.x * 8 + (threadIdx.x >> 5);
    const int rowTile = tile >> 3;  // HID_F/16 == 8 column tiles
    const int colTile = tile & 7;
    const int rowBase = rowTile << 4;
    if (rowBase >= N_NODES) return;  // wave-uniform: EXEC stays all-ones
    const int l15 = lane & 15;
    const int kOff = (lane >> 4) << 1;  // K sub-offset: 0 (lanes 0-15) or 2 (16-31)
    const int row = min(rowBase + l15, N_NODES - 1);
    const int col = (colTile << 4) + l15;
    const float dinv = 1.0f / fmaxf(deg[row], 1.0f);
    const float* aRow = agg + (size_t)row * IN_F;
    const float* xRow = x + (size_t)row * IN_F;
    const float* wl = W1l + (size_t)col * IN_F;  // B[k,n] = W1l[n,k]
    const float* wr = W1r + (size_t)col * IN_F;
    v8f c = {};
#pragma unroll
    for (int k = 0; k < IN_F; k += 4) {
        v2f aM, aX, bL, bR;
        aM.x = aRow[k + kOff] * dinv; aM.y = aRow[k + kOff + 1] * dinv;
        aX.x = xRow[k + kOff];        aX.y = xRow[k + kOff + 1];
        bL.x = wl[k + kOff];          bL.y = wl[k + kOff + 1];
        bR.x = wr[k + kOff];          bR.y = wr[k + kOff + 1];
        c = __builtin_amdgcn_wmma_f32_16x16x4_f32(false, aM, false, bL,
                                                  (short)0, c, false, false);
        c = __builtin_amdgcn_wmma_f32_16x16x4_f32(false, aX, false, bR,
                                                  (short)0, c, false, false);
    }
    const float bias = b1[col];
#pragma unroll
    for (int v = 0; v < 8; ++v) {
        int r = rowBase + v + ((lane >> 4) << 3);  // D: VGPR v -> row v (+8 for hi half)
        if (r < N_NODES)
            h[(size_t)r * HID_F + col] = fmaxf(c[v] + bias, 0.0f);
    }
}

// ---------------- layer 2: out = mean2@W2l.T + h@W2r.T + b2 ----------------
__global__ void __launch_bounds__(256) sage_layer2(
    const float* __restrict__ h, const float* __restrict__ agg,
    const float* __restrict__ deg, const float* __restrict__ W2l,
    const float* __restrict__ W2r, const float* __restrict__ b2,
    float* __restrict__ out) {
    const int lane = threadIdx.x & 31;
    const int tile = blockIdx.x * 8 + (threadIdx.x >> 5);
    const int rowTile = tile >> 3;  // ceil(121/16) == 8 column tiles
    const int colTile = tile & 7;
    const int rowBase = rowTile << 4;
    if (rowBase >= N_NODES) return;
    const int l15 = lane & 15;
    const int kOff = (lane >> 4) << 1;
    const int row = min(rowBase + l15, N_NODES - 1);
    const int colAbs = (colTile << 4) + l15;
    const int col = min(colAbs, OUT_F - 1);  // clamp loads in partial last tile
    const float dinv = 1.0f / fmaxf(deg[row], 1.0f);
    const float* aRow = agg + (size_t)row * HID_F;
    const float* hRow = h + (size_t)row * HID_F;
    const float* wl = W2l + (size_t)col * HID_F;
    const float* wr = W2r + (size_t)col * HID_F;
    v8f c = {};
#pragma unroll
    for (int k = 0; k < HID_F; k += 4) {
        v2f aM, aX, bL, bR;
        aM.x = aRow[k + kOff] * dinv; aM.y = aRow[k + kOff + 1] * dinv;
        aX.x = hRow[k + kOff];        aX.y = hRow[k + kOff + 1];
        bL.x = wl[k + kOff];          bL.y = wl[k + kOff + 1];
        bR.x = wr[k + kOff];          bR.y = wr[k + kOff + 1];
        c = __builtin_amdgcn_wmma_f32_16x16x4_f32(false, aM, false, bL,
                                                  (short)0, c, false, false);
        c = __builtin_amdgcn_wmma_f32_16x16x4_f32(false, aX, false, bR,
                                                  (short)0, c, false, false);
    }
    const float bias = b2[col];
#pragma unroll
    for (int v = 0; v < 8; ++v) {
        int r = rowBase + v + ((lane >> 4) << 3);
        if (r < N_NODES && colAbs < OUT_F)
            out[(size_t)r * OUT_F + colAbs] = c[v] + bias;
    }
}

extern "C" void kernel_launch(void* const* d_in, const int* in_sizes, int n_in,
                              void* d_out, int out_size, void* d_ws, size_t ws_size,
                              hipStream_t stream) {
    const float* x = (const float*)d_in[0];
    const long long* ei = (const long long*)d_in[1];  // int64 [2, E]
    const float* W1l = (const float*)d_in[2];
    const float* W1r = (const float*)d_in[3];
    const float* b1 = (const float*)d_in[4];
    const float* W2l = (const float*)d_in[5];
    const float* W2r = (const float*)d_in[6];
    const float* b2 = (const float*)d_in[7];
    const long long* src = ei;
    const long long* dst = ei + N_EDGES;
    float* out = (float*)d_out;

    // Workspace layout (contiguous so one zero pass covers all accumulators):
    // agg1 [N,64] | agg2 [N,128] | deg [N] | h [N,128]
    char* ws = (char*)d_ws;
    float* agg1 = (float*)ws;
    float* agg2 = (float*)(ws + (size_t)N_NODES * IN_F * 4);
    float* deg = (float*)(ws + (size_t)N_NODES * (IN_F + HID_F) * 4);
    float* h = (float*)(ws + (size_t)N_NODES * (IN_F + HID_F + 1) * 4);

    const int nzero = N_NODES * (IN_F + HID_F + 1);  // agg1+agg2+deg
    zero_f32<<<2048, 256, 0, stream>>>(agg1, nzero);

    degree_kernel<<<(N_EDGES + 255) / 256, 256, 0, stream>>>(dst, deg, N_EDGES);

    scatter_add<6><<<(N_EDGES * IN_F) / 256, 256, 0, stream>>>(x, src, dst,
                                                               agg1, N_EDGES);

    // 6250 row tiles x 8 col tiles, 8 waves (tiles) per 256-thread block
    sage_layer1<<<N_NODES / 16, 256, 0, stream>>>(x, agg1, deg, W1l, W1r, b1, h);

    scatter_add<7><<<(N_EDGES * HID_F) / 256, 256, 0, stream>>>(h, src, dst,
                                                                agg2, N_EDGES);

    sage_layer2<<<N_NODES / 16, 256, 0, stream>>>(h, agg2, deg, W2l, W2r, b2, out);
}